// Contrast_59794534695372
// MI455X (gfx1250) — compile-verified
//
#include <hip/hip_runtime.h>
#include <hip/hip_bf16.h>
#include <math.h>

// Problem constants (match reference)
#define TAUF   0.07f
#define LP     4
#define BP     4
#define CP     16
#define NP     (64*64*64)
#define SP     50
#define SPAD   64
#define LCHW   17   // padded LDS row stride (floats) to avoid bank conflicts

typedef __attribute__((ext_vector_type(2))) float v2f;
typedef __attribute__((ext_vector_type(8))) float v8f;

// ---------------------------------------------------------------------------
// Kernel 1: gather sampled voxels, L2-normalize channel vectors, compute
// positive-pair term and self (diagonal) term. One thread per (b, s).
// ---------------------------------------------------------------------------
__global__ void k_gather(const float* __restrict__ proj,
                         const int*   __restrict__ indices,
                         const int*   __restrict__ idxp,
                         float* __restrict__ chat,   // [B][64][16], rows >=50 zero
                         float* __restrict__ invn,   // [B][64] 1/max(||chat||,1e-8)
                         float* __restrict__ pose,   // [B][64] exp(pos_sim/tau)
                         float* __restrict__ selfe)  // [B][64] exp(cos_jj/tau)
{
    int b = blockIdx.x;
    int s = threadIdx.x;            // blockDim = 64
    int row = b * SPAD + s;

    if (s >= SP) {                  // zero-pad rows 50..63
#pragma unroll
        for (int c = 0; c < CP; ++c) chat[row * CP + c] = 0.0f;
        invn[row] = 0.0f; pose[row] = 0.0f; selfe[row] = 0.0f;
        return;
    }

    const int idx = idxp[0];
    const int n   = indices[b * SP + s];

    // current projection vector (channel stride = N)
    float v[CP];
    float nrm2 = 0.0f;
    const float* pc = proj + (size_t)((idx * BP + b) * CP) * (size_t)NP + (size_t)n;
#pragma unroll
    for (int c = 0; c < CP; ++c) {
        float x = pc[(size_t)c * (size_t)NP];
        v[c] = x; nrm2 += x * x;
    }
    float nrm = sqrtf(nrm2);
    float s1  = 1.0f / fmaxf(nrm, 1e-12f);      // F.normalize eps
#pragma unroll
    for (int c = 0; c < CP; ++c) chat[row * CP + c] = v[c] * s1;

    float cn = nrm * s1;                        // ||chat|| (1 or ~0)
    float ci = 1.0f / fmaxf(cn, 1e-8f);         // cosine_similarity eps
    invn[row]  = ci;
    selfe[row] = expf((cn * cn * ci * ci) / TAUF);   // exp(cos_jj / tau)

    // positive term: sum of cosines against the other L-1 projections
    float psum = 0.0f;
    for (int l = 0; l < LP; ++l) {
        if (l == idx) continue;
        const float* pp = proj + (size_t)((l * BP + b) * CP) * (size_t)NP + (size_t)n;
        float pv[CP];
        float pn2 = 0.0f;
#pragma unroll
        for (int c = 0; c < CP; ++c) {
            float x = pp[(size_t)c * (size_t)NP];
            pv[c] = x; pn2 += x * x;
        }
        float pn  = sqrtf(pn2);
        float sp  = 1.0f / fmaxf(pn, 1e-12f);
        float pcn = pn * sp;
        float pci = 1.0f / fmaxf(pcn, 1e-8f);
        float dot = 0.0f;
#pragma unroll
        for (int c = 0; c < CP; ++c) dot += (v[c] * s1) * (pv[c] * sp);
        psum += dot * ci * pci;
    }
    pose[row] = expf(psum / TAUF);
}

// ---------------------------------------------------------------------------
// Kernel 2: per-batch 64x64 (padded) Gram matrix via V_WMMA_F32_16X16X4_F32,
// then exp(cos/tau) column sums, diagonal subtraction, per-batch loss.
// One wave32 per batch; EXEC all-ones at every WMMA.
// ---------------------------------------------------------------------------
__global__ void k_gram(const float* __restrict__ chat,
                       const float* __restrict__ invn,
                       const float* __restrict__ pose,
                       const float* __restrict__ selfe,
                       float* __restrict__ bloss)   // [B]
{
    __shared__ float lch[SPAD * LCHW];
    __shared__ float linv[SPAD];
    __shared__ float colsum[SPAD];
    __shared__ float lloss[SPAD];

    const int b    = blockIdx.x;
    const int lane = threadIdx.x;     // 0..31

    // stage padded Chat [64x16] into LDS (row stride 17)
    for (int i = lane; i < SPAD * CP; i += 32) {
        int r = i >> 4, c = i & 15;
        lch[r * LCHW + c] = chat[(b * SPAD + r) * CP + c];
    }
    linv[lane]        = invn[b * SPAD + lane];
    linv[lane + 32]   = invn[b * SPAD + lane + 32];
    colsum[lane]      = 0.0f;
    colsum[lane + 32] = 0.0f;
    __syncthreads();

    // FP32 WMMA 16x16x4 operand layout:
    //  A (16x4): lanes 0-15 hold (K0,K1) of row M=lane; lanes 16-31 hold (K2,K3)
    //  B (4x16): VGPR0/1 = K rows striped over N lanes -> same gather pattern
    const int mrow  = lane & 15;
    const int hk    = (lane >= 16) ? 2 : 0;
    const int rhalf = (lane >= 16) ? 8 : 0;

    for (int ti = 0; ti < 4; ++ti) {
        v2f av[4];
        const int arow = ti * 16 + mrow;
#pragma unroll
        for (int kk = 0; kk < 4; ++kk) {
            av[kk].x = lch[arow * LCHW + kk * 4 + hk];
            av[kk].y = lch[arow * LCHW + kk * 4 + hk + 1];
        }
        for (int tj = 0; tj < 4; ++tj) {
            v2f bv[4];
            const int brow = tj * 16 + mrow;
#pragma unroll
            for (int kk = 0; kk < 4; ++kk) {
                bv[kk].x = lch[brow * LCHW + kk * 4 + hk];
                bv[kk].y = lch[brow * LCHW + kk * 4 + hk + 1];
            }
            v8f acc = {0.0f, 0.0f, 0.0f, 0.0f, 0.0f, 0.0f, 0.0f, 0.0f};
#pragma unroll
            for (int kk = 0; kk < 4; ++kk) {
                acc = __builtin_amdgcn_wmma_f32_16x16x4_f32(
                    /*neg_a=*/false, av[kk],
                    /*neg_b=*/false, bv[kk],
                    /*c_mod=*/(short)0, acc,
                    /*reuse_a=*/false, /*reuse_b=*/false);
            }
            // acc[v] = D[M = ti*16 + v + rhalf][N = tj*16 + mrow]
            const int   gcol = tj * 16 + mrow;
            const float icol = linv[gcol];
            float partial = 0.0f;
            const int rbase = ti * 16 + rhalf;
#pragma unroll
            for (int vv = 0; vv < 8; ++vv) {
                int grow = rbase + vv;
                if (grow < SP && gcol < SP)
                    partial += expf((acc[vv] * linv[grow] * icol) / TAUF);
            }
            if (gcol < SP) atomicAdd(&colsum[gcol], partial);  // ds_add_f32
        }
    }
    __syncthreads();

    // neg[s] = colsum[s] - diag; loss_s = -log(pe / (pe + neg + 1e-8))
    for (int s = lane; s < SPAD; s += 32) {
        float ls = 0.0f;
        if (s < SP) {
            float neg = colsum[s] - selfe[b * SPAD + s];
            float pe  = pose[b * SPAD + s];
            ls = -logf(pe / (pe + neg + 1e-8f));
        }
        lloss[s] = ls;
    }
    __syncthreads();
    if (lane == 0) {
        float t = 0.0f;
        for (int s = 0; s < SP; ++s) t += lloss[s];
        bloss[b] = t / (float)SP;
    }
}

// ---------------------------------------------------------------------------
// Kernel 3: mean over batches -> scalar loss
// ---------------------------------------------------------------------------
__global__ void k_final(const float* __restrict__ bloss, float* __restrict__ out)
{
    if (blockIdx.x == 0 && threadIdx.x == 0) {
        float t = 0.0f;
        for (int b = 0; b < BP; ++b) t += bloss[b];
        out[0] = t / (float)BP;
    }
}

// ---------------------------------------------------------------------------
extern "C" void kernel_launch(void* const* d_in, const int* in_sizes, int n_in,
                              void* d_out, int out_size, void* d_ws, size_t ws_size,
                              hipStream_t stream)
{
    const float* proj    = (const float*)d_in[0];
    // d_in[1] = mask (unused by the loss itself)
    const int*   indices = (const int*)d_in[2];
    const int*   idxp    = (const int*)d_in[3];

    float* ws    = (float*)d_ws;
    float* chat  = ws;                       // B*64*16 = 4096 floats
    float* invn  = chat  + BP * SPAD * CP;   // B*64
    float* pose  = invn  + BP * SPAD;        // B*64
    float* selfe = pose  + BP * SPAD;        // B*64
    float* bloss = selfe + BP * SPAD;        // B

    k_gather<<<BP, SPAD, 0, stream>>>(proj, indices, idxp, chat, invn, pose, selfe);
    k_gram  <<<BP, 32,   0, stream>>>(chat, invn, pose, selfe, bloss);
    k_final <<<1,  1,    0, stream>>>(bloss, (float*)d_out);
}